// MultiHeadedAttention_17489106830115
// MI455X (gfx1250) — compile-verified
//
#include <hip/hip_runtime.h>
#include <stddef.h>

// ---------------------------------------------------------------------------
// MultiHeadedAttention (no softmax) on gfx1250 via V_WMMA_F32_16X16X4_F32.
// out = (Q Kᵀ · scale) V == Q · (scale · Kᵀ V)     (associativity, no softmax)
// Phase 1: per (b,h,chunk of 128 rows): K,V tiles via WMMA GEMM over C=1024
//          (double-buffered LDS pipeline), then partial M = scale·KᵀV (64x64).
// Phase 2: deterministic reduction of chunk partials -> M[bh][64][64].
// Phase 3: Q tiles via WMMA GEMM, then out = Q @ M via WMMA.
// All LDS tiles laid out so every WMMA fragment is a contiguous 8-byte pair
// (single ds_load_b64, no register repacking), conflict-free padded strides.
// Workspace: (4*16*16 + 4*16) * 64*64 floats = ~17.8 MB.
// ---------------------------------------------------------------------------

#define B_SZ   4
#define N_SEQ  2048
#define C_IN   1024
#define NH     16
#define DH     64
#define W_COLS (3 * C_IN)          // 3072
#define ROWS   128                 // n-rows per workgroup tile
#define NCHUNK (N_SEQ / ROWS)      // 16
#define BK     16                  // k-block staged in LDS
#define NKB    (C_IN / BK)         // 64
#define QKV_SCALE 32.0f            // sqrt(1024)

typedef __attribute__((ext_vector_type(2))) float v2f;
typedef __attribute__((ext_vector_type(8))) float v8f;

__device__ __forceinline__ v8f wmma4(v2f a, v2f b, v8f c) {
  // D = A(16x4) * B(4x16) + C(16x16), fp32 WMMA
  return __builtin_amdgcn_wmma_f32_16x16x4_f32(false, a, false, b, (short)0, c,
                                               false, false);
}
__device__ __forceinline__ v8f vzero8() {
  v8f z;
  #pragma unroll
  for (int i = 0; i < 8; ++i) z[i] = 0.0f;
  return z;
}
__device__ __forceinline__ v2f ldpair(const float* p) {
  v2f r; r.x = p[0]; r.y = p[1]; return r;
}

// ---------------------------------------------------------------------------
// Phase 1: partial M = scale * K_chunkᵀ V_chunk
// grid.x = B_SZ*NH*NCHUNK, block = 256 (8 waves)
// GEMM1 partition: wave -> 32-row block (rb=w>>1) x 32-col block (cb=w&1)
// of both K and V: 8 accumulator tiles, 6 b64 LDS loads per 8 WMMAs.
// ---------------------------------------------------------------------------
__global__ __launch_bounds__(256)
void kv_to_m_partial(const float* __restrict__ x, const float* __restrict__ Wqkv,
                     const float* __restrict__ bias, float* __restrict__ Mpart) {
  __shared__ float lx [2][ROWS][20];   // x tile, k fastest (20.5 KB)
  __shared__ float lwt[2][192][20];    // W tile TRANSPOSED [col][k] (30.7 KB)
  __shared__ float lkT[DH][132];       // K parked transposed [j][i] (33.8 KB)
  __shared__ float lvT[DH][132];       // V parked transposed [v][i] (33.8 KB)

  const int tid  = threadIdx.x;
  const int lane = tid & 31;
  const int wv   = tid >> 5;
  const int lo   = lane & 15;
  const int hi   = lane >> 4;          // 0: K=0,1 half, 1: K=2,3 half
  const int rb   = wv >> 1;            // row block: rows 32*rb .. +31
  const int cb   = wv & 1;             // col block: dims 32*cb .. +31

  const int chunk = blockIdx.x % NCHUNK;
  const int bh    = blockIdx.x / NCHUNK;
  const int h     = bh % NH;
  const int b     = bh / NH;

  const float* xb = x + ((size_t)b * N_SEQ + (size_t)chunk * ROWS) * C_IN;
  const float* wh = Wqkv + (size_t)h * 192;   // 192 = 64 dims * {q,k,v}

  v8f aK[2][2], aV[2][2];
  #pragma unroll
  for (int rt = 0; rt < 2; ++rt)
    #pragma unroll
    for (int ct = 0; ct < 2; ++ct) { aK[rt][ct] = vzero8(); aV[rt][ct] = vzero8(); }

  // --- staging registers for the double-buffered pipeline ------------------
  float4 rx[2];
  float  rw[3][4];

  auto load_stage = [&](int k0) {
    #pragma unroll
    for (int i = 0; i < 2; ++i) {                 // x: 512 float4s
      int e = tid + i * 256, r = e >> 2, c = (e & 3) * 4;
      rx[i] = *reinterpret_cast<const float4*>(&xb[(size_t)r * C_IN + k0 + c]);
    }
    #pragma unroll
    for (int i = 0; i < 3; ++i) {                 // W: 768 (col, kq) tasks
      int e = tid + i * 256, col = e % 192, kq = e / 192;
      #pragma unroll
      for (int j = 0; j < 4; ++j)                 // coalesced across lanes
        rw[i][j] = wh[(size_t)(k0 + kq * 4 + j) * W_COLS + col];
    }
  };
  auto store_stage = [&](int buf) {
    #pragma unroll
    for (int i = 0; i < 2; ++i) {
      int e = tid + i * 256, r = e >> 2, c = (e & 3) * 4;
      *reinterpret_cast<float4*>(&lx[buf][r][c]) = rx[i];
    }
    #pragma unroll
    for (int i = 0; i < 3; ++i) {
      int e = tid + i * 256, col = e % 192, kq = e / 192;
      float4 v4 = make_float4(rw[i][0], rw[i][1], rw[i][2], rw[i][3]);
      *reinterpret_cast<float4*>(&lwt[buf][col][kq * 4]) = v4;   // transpose
    }
  };
  auto compute = [&](int buf) {
    #pragma unroll
    for (int ks = 0; ks < BK; ks += 4) {
      v2f a[2], bK[2], bV[2];
      #pragma unroll
      for (int rt = 0; rt < 2; ++rt)
        a[rt] = ldpair(&lx[buf][rb * 32 + rt * 16 + lo][ks + 2 * hi]);
      #pragma unroll
      for (int ct = 0; ct < 2; ++ct) {
        const int j = cb * 32 + ct * 16 + lo;
        bK[ct] = ldpair(&lwt[buf][j * 3 + 1][ks + 2 * hi]);
        bV[ct] = ldpair(&lwt[buf][j * 3 + 2][ks + 2 * hi]);
      }
      #pragma unroll
      for (int rt = 0; rt < 2; ++rt)
        #pragma unroll
        for (int ct = 0; ct < 2; ++ct) {
          aK[rt][ct] = wmma4(a[rt], bK[ct], aK[rt][ct]);
          aV[rt][ct] = wmma4(a[rt], bV[ct], aV[rt][ct]);
        }
    }
  };

  // --- pipelined main loop -------------------------------------------------
  load_stage(0);
  store_stage(0);
  for (int kb = 0; kb < NKB; ++kb) {
    if (kb + 1 < NKB) load_stage((kb + 1) * BK);   // overlap with compute
    __syncthreads();                               // buf[kb&1] ready
    compute(kb & 1);
    if (kb + 1 < NKB) store_stage((kb + 1) & 1);   // safe: one barrier/iter
  }

  // --- add bias, park K,V transposed in LDS --------------------------------
  #pragma unroll
  for (int ct = 0; ct < 2; ++ct) {
    const int j  = cb * 32 + ct * 16 + lo;
    const float bk = bias[(h * DH + j) * 3 + 1];
    const float bv = bias[(h * DH + j) * 3 + 2];
    #pragma unroll
    for (int rt = 0; rt < 2; ++rt) {
      const int i0 = rb * 32 + rt * 16 + hi * 8;
      #pragma unroll
      for (int r = 0; r < 8; ++r) {
        lkT[j][i0 + r] = aK[rt][ct][r] + bk;
        lvT[j][i0 + r] = aV[rt][ct][r] + bv;
      }
    }
  }
  __syncthreads();

  // --- M_part[j][v] = sum_i K[i][j]*V[i][v]; both frags contiguous b64 -----
  v8f mA = vzero8(), mB = vzero8();
  const int jt = wv >> 1;
  const int vt = (wv & 1) * 2;
  for (int ii = 0; ii < ROWS; ii += 4) {
    v2f a  = ldpair(&lkT[16 * jt + lo][ii + 2 * hi]);       // A[M=j][K=i]
    v2f b0 = ldpair(&lvT[16 * vt + lo][ii + 2 * hi]);       // B[K=i][N=v]
    mA = wmma4(a, b0, mA);
    v2f b1 = ldpair(&lvT[16 * (vt + 1) + lo][ii + 2 * hi]);
    mB = wmma4(a, b1, mB);
  }

  float* mp = Mpart + ((size_t)bh * NCHUNK + chunk) * (DH * DH);
  #pragma unroll
  for (int r = 0; r < 8; ++r) {
    const int j = 16 * jt + hi * 8 + r;
    mp[j * DH + 16 * vt + lo]       = mA[r] * QKV_SCALE;
    mp[j * DH + 16 * (vt + 1) + lo] = mB[r] * QKV_SCALE;
  }
}

// ---------------------------------------------------------------------------
// Phase 2: M[bh] = sum over NCHUNK partials (deterministic)
// ---------------------------------------------------------------------------
__global__ __launch_bounds__(256)
void reduce_m(const float* __restrict__ Mpart, float* __restrict__ M) {
  const int bh = blockIdx.x;
  for (int e = threadIdx.x; e < DH * DH; e += 256) {
    float s = 0.0f;
    #pragma unroll
    for (int c = 0; c < NCHUNK; ++c)
      s += Mpart[((size_t)bh * NCHUNK + c) * (DH * DH) + e];
    M[(size_t)bh * (DH * DH) + e] = s;
  }
}

// ---------------------------------------------------------------------------
// Phase 3: out[b,h, rows, :] = (X·Wq + bq) @ M[bh]
// ---------------------------------------------------------------------------
__global__ __launch_bounds__(256)
void q_matmul_out(const float* __restrict__ x, const float* __restrict__ Wqkv,
                  const float* __restrict__ bias, const float* __restrict__ Mfull,
                  float* __restrict__ out) {
  __shared__ float lx [2][ROWS][20];
  __shared__ float lwt[2][192][20];    // transposed [col][k]
  __shared__ float lq [ROWS][68];      // Q row-major [i][j] (A-frags contiguous)
  __shared__ float lmT[DH][68];        // M transposed [d][j] (B-frags contiguous)

  const int tid  = threadIdx.x;
  const int lane = tid & 31;
  const int wv   = tid >> 5;
  const int lo   = lane & 15;
  const int hi   = lane >> 4;
  const int rb   = wv >> 1;
  const int cb   = wv & 1;

  const int chunk = blockIdx.x % NCHUNK;
  const int bh    = blockIdx.x / NCHUNK;
  const int h     = bh % NH;
  const int b     = bh / NH;

  const float* xb = x + ((size_t)b * N_SEQ + (size_t)chunk * ROWS) * C_IN;
  const float* wh = Wqkv + (size_t)h * 192;

  // preload M[bh] transposed: lmT[d][j] = M[j][d]
  {
    const float* msrc = Mfull + (size_t)bh * (DH * DH);
    for (int e = tid; e < DH * DH; e += 256)
      lmT[e & 63][e >> 6] = msrc[e];
  }

  v8f aQ[2][2];
  #pragma unroll
  for (int rt = 0; rt < 2; ++rt)
    #pragma unroll
    for (int ct = 0; ct < 2; ++ct) aQ[rt][ct] = vzero8();

  float4 rx[2];
  float  rw[3][4];

  auto load_stage = [&](int k0) {
    #pragma unroll
    for (int i = 0; i < 2; ++i) {
      int e = tid + i * 256, r = e >> 2, c = (e & 3) * 4;
      rx[i] = *reinterpret_cast<const float4*>(&xb[(size_t)r * C_IN + k0 + c]);
    }
    #pragma unroll
    for (int i = 0; i < 3; ++i) {
      int e = tid + i * 256, col = e % 192, kq = e / 192;
      #pragma unroll
      for (int j = 0; j < 4; ++j)
        rw[i][j] = wh[(size_t)(k0 + kq * 4 + j) * W_COLS + col];
    }
  };
  auto store_stage = [&](int buf) {
    #pragma unroll
    for (int i = 0; i < 2; ++i) {
      int e = tid + i * 256, r = e >> 2, c = (e & 3) * 4;
      *reinterpret_cast<float4*>(&lx[buf][r][c]) = rx[i];
    }
    #pragma unroll
    for (int i = 0; i < 3; ++i) {
      int e = tid + i * 256, col = e % 192, kq = e / 192;
      float4 v4 = make_float4(rw[i][0], rw[i][1], rw[i][2], rw[i][3]);
      *reinterpret_cast<float4*>(&lwt[buf][col][kq * 4]) = v4;
    }
  };
  auto compute = [&](int buf) {
    #pragma unroll
    for (int ks = 0; ks < BK; ks += 4) {
      v2f a[2], bQ[2];
      #pragma unroll
      for (int rt = 0; rt < 2; ++rt)
        a[rt] = ldpair(&lx[buf][rb * 32 + rt * 16 + lo][ks + 2 * hi]);
      #pragma unroll
      for (int ct = 0; ct < 2; ++ct) {
        const int j = cb * 32 + ct * 16 + lo;
        bQ[ct] = ldpair(&lwt[buf][j * 3][ks + 2 * hi]);      // Q column sel=0
      }
      #pragma unroll
      for (int rt = 0; rt < 2; ++rt)
        #pragma unroll
        for (int ct = 0; ct < 2; ++ct)
          aQ[rt][ct] = wmma4(a[rt], bQ[ct], aQ[rt][ct]);
    }
  };

  load_stage(0);
  store_stage(0);
  for (int kb = 0; kb < NKB; ++kb) {
    if (kb + 1 < NKB) load_stage((kb + 1) * BK);
    __syncthreads();
    compute(kb & 1);
    if (kb + 1 < NKB) store_stage((kb + 1) & 1);
  }

  // add bias, park Q row-major
  #pragma unroll
  for (int ct = 0; ct < 2; ++ct) {
    const int j  = cb * 32 + ct * 16 + lo;
    const float bq = bias[(h * DH + j) * 3 + 0];
    #pragma unroll
    for (int rt = 0; rt < 2; ++rt) {
      const int i0 = rb * 32 + rt * 16 + hi * 8;
      #pragma unroll
      for (int r = 0; r < 8; ++r)
        lq[i0 + r][j] = aQ[rt][ct][r] + bq;
    }
  }
  __syncthreads();

  // out_tile[128][64] = lq @ M ; wave -> rows 16w, all 4 col tiles
  v8f o[4];
  #pragma unroll
  for (int t = 0; t < 4; ++t) o[t] = vzero8();

  for (int kk = 0; kk < DH; kk += 4) {
    v2f a = ldpair(&lq[wv * 16 + lo][kk + 2 * hi]);          // A[M=i][K=j]
    #pragma unroll
    for (int t = 0; t < 4; ++t) {
      v2f bb = ldpair(&lmT[16 * t + lo][kk + 2 * hi]);       // B[K=j][N=d]
      o[t] = wmma4(a, bb, o[t]);
    }
  }

  // out layout [b, h, n, d]
  float* ob = out + ((size_t)bh * N_SEQ + (size_t)chunk * ROWS + wv * 16) * DH;
  #pragma unroll
  for (int t = 0; t < 4; ++t)
    #pragma unroll
    for (int r = 0; r < 8; ++r)
      ob[(size_t)(hi * 8 + r) * DH + 16 * t + lo] = o[t][r];
}

// ---------------------------------------------------------------------------
extern "C" void kernel_launch(void* const* d_in, const int* in_sizes, int n_in,
                              void* d_out, int out_size, void* d_ws, size_t ws_size,
                              hipStream_t stream) {
  const float* x    = (const float*)d_in[0];   // [4, 2048, 1024]
  const float* W    = (const float*)d_in[1];   // [1024, 3072]
  const float* bqkv = (const float*)d_in[2];   // [3072]
  float* out = (float*)d_out;                  // [4, 16, 2048, 64]

  float* Mpart = (float*)d_ws;                                   // [64][16][4096]
  float* M     = Mpart + (size_t)B_SZ * NH * NCHUNK * DH * DH;   // [64][4096]

  kv_to_m_partial<<<B_SZ * NH * NCHUNK, 256, 0, stream>>>(x, W, bqkv, Mpart);
  reduce_m<<<B_SZ * NH, 256, 0, stream>>>(Mpart, M);
  q_matmul_out<<<B_SZ * NH * NCHUNK, 256, 0, stream>>>(x, W, bqkv, M, out);

  (void)in_sizes; (void)n_in; (void)out_size; (void)ws_size;
}